// Sampler_74165495267838
// MI455X (gfx1250) — compile-verified
//
#include <hip/hip_runtime.h>
#include <hip/hip_bf16.h>
#include <math.h>

// Sampler: penalties + temperature + top-k/top-p/top-a + softmax, WITHOUT sorting.
// S=256 rows, V=128000. Thresholds found by (weighted) radix select on ordered keys.

#define NSEQ  256
#define VOCAB 128000
#define HISTN 200
#define TPB   256
#define VEC   4
#define TILE  (TPB*VEC)       // 1024
#define NITER (VOCAB/TILE)    // 125 (exact)
#define NBIN  256

typedef __attribute__((ext_vector_type(2))) float v2f;
typedef __attribute__((ext_vector_type(8))) float v8f;

// monotonic uint key: larger float -> larger key
__device__ __forceinline__ unsigned fkey(float x){
  unsigned u = __float_as_uint(x);
  return (u & 0x80000000u) ? ~u : (u | 0x80000000u);
}

// Wave32 sum via the matrix pipe: A(16x4)=lane partials in K0/K2 columns,
// B(4x16)=ones -> D rows are pairwise lane sums; fold 8 acc VGPRs + xor-16.
__device__ __forceinline__ float wave_sum_wmma(float x){
  v2f a; a[0] = x;   a[1] = 0.0f;
  v2f b; b[0] = 1.0f; b[1] = 1.0f;
  v8f c = {0.f,0.f,0.f,0.f,0.f,0.f,0.f,0.f};
  c = __builtin_amdgcn_wmma_f32_16x16x4_f32(false, a, false, b, (short)0, c, false, false);
  float s = c[0]+c[1]+c[2]+c[3]+c[4]+c[5]+c[6]+c[7];
  s += __shfl_xor(s, 16, 32);
  return s;
}

__device__ __forceinline__ float block_sum(float x, float* s_red){
  float w = wave_sum_wmma(x);
  int tid = threadIdx.x;
  if ((tid & 31) == 0) s_red[tid >> 5] = w;
  __syncthreads();
  float tot = 0.f;
#pragma unroll
  for (int i = 0; i < TPB/32; i++) tot += s_red[i];
  __syncthreads();
  return tot;
}

// ---------------- K1: penalties + temperature, row max, level-0 histogram ----
__global__ __launch_bounds__(TPB) void k1_penalize(
    const float* __restrict__ logits, const int* __restrict__ toks,
    const float* __restrict__ presence, const float* __restrict__ frequency,
    const float* __restrict__ temps, float* __restrict__ xout,
    float* __restrict__ ws_max, unsigned* __restrict__ ws_hist)
{
  const int row = blockIdx.x, tid = threadIdx.x;
  __shared__ unsigned s_bm[VOCAB/32];     // 16 KB token-presence bitmap
  __shared__ int      s_tok[HISTN];
  __shared__ unsigned s_h[NBIN];
  __shared__ float    s_red[TPB/32];
  __shared__ float    s_stage[2][TILE];   // async double buffer (8 KB)

  for (int i = tid; i < VOCAB/32; i += TPB) s_bm[i] = 0u;
  s_h[tid] = 0u;
  if (tid < HISTN) s_tok[tid] = toks[row*HISTN + tid];
  __syncthreads();
  if (tid < HISTN) { int t = s_tok[tid]; atomicOr(&s_bm[(unsigned)t >> 5], 1u << (t & 31)); }
  __syncthreads();

  const float fp = frequency[row], pp = presence[row], invT = 1.0f/temps[row];
  const float* lrow = logits + (size_t)row*VOCAB;
  float*       orow = xout   + (size_t)row*VOCAB;
  float lmax = -INFINITY;

  // prologue: async-copy tile 0 into LDS (each lane loads its own 16B slot)
  {
    unsigned lo = (unsigned)(size_t)&s_stage[0][tid*VEC];
    unsigned long long ga = (unsigned long long)(size_t)(lrow + tid*VEC);
    asm volatile("global_load_async_to_lds_b128 %0, %1, off" :: "v"(lo), "v"(ga) : "memory");
  }
  for (int it = 0; it < NITER; it++) {
    if (it + 1 < NITER) {
      unsigned lo = (unsigned)(size_t)&s_stage[(it+1)&1][tid*VEC];
      unsigned long long ga = (unsigned long long)(size_t)(lrow + (size_t)(it+1)*TILE + tid*VEC);
      asm volatile("global_load_async_to_lds_b128 %0, %1, off" :: "v"(lo), "v"(ga) : "memory");
      asm volatile("s_wait_asynccnt 1" ::: "memory");   // tile `it` landed (in-order)
    } else {
      asm volatile("s_wait_asynccnt 0" ::: "memory");
    }
    const int base = it*TILE + tid*VEC;
    float4 v = *(float4*)&s_stage[it & 1][tid*VEC];
    float e[4] = {v.x, v.y, v.z, v.w};
#pragma unroll
    for (int j = 0; j < VEC; j++) {
      int t = base + j;
      float x = e[j];
      if ((s_bm[(unsigned)t >> 5] >> (t & 31)) & 1u) {   // rare path (<=200/row)
        int c = 0;
        for (int i = 0; i < HISTN; i++) c += (s_tok[i] == t) ? 1 : 0;
        x -= fp*(float)c + pp;
      }
      x *= invT;
      lmax = fmaxf(lmax, x);
      atomicAdd(&s_h[fkey(x) >> 24], 1u);
      e[j] = x;
    }
    float4 o; o.x = e[0]; o.y = e[1]; o.z = e[2]; o.w = e[3];
    *(float4*)(orow + base) = o;
  }
#pragma unroll
  for (int o = 16; o > 0; o >>= 1) lmax = fmaxf(lmax, __shfl_xor(lmax, o, 32));
  if ((tid & 31) == 0) s_red[tid >> 5] = lmax;
  __syncthreads();
  if (tid == 0) {
    float mm = s_red[0];
    for (int i = 1; i < TPB/32; i++) mm = fmaxf(mm, s_red[i]);
    ws_max[row] = mm;
  }
  ws_hist[row*NBIN + tid] = s_h[tid];
}

// ---------------- K2: radix selects -> per-row thresholds + Z' --------------
__global__ __launch_bounds__(TPB) void k2_select(
    const float* __restrict__ x, const float* __restrict__ ws_max,
    const unsigned* __restrict__ ws_hist, const int* __restrict__ top_ks,
    const float* __restrict__ top_ps, const float* __restrict__ top_as,
    float* __restrict__ ws_res)
{
  const int row = blockIdx.x, tid = threadIdx.x;
  const float* xr = x + (size_t)row*VOCAB;
  __shared__ unsigned sb[NBIN];
  __shared__ float    swb[NBIN];
  __shared__ float    s_red[TPB/32];
  __shared__ unsigned s_prefix;
  __shared__ int      s_rem;
  __shared__ float    s_trem;
  __shared__ int      s_hasP;

  const float m = ws_max[row];
  int k = top_ks[row]; k = k < 1 ? 1 : (k > VOCAB ? VOCAB : k);

  // ---- exact key of the k-th largest (4-level radix select) ----
  sb[tid] = ws_hist[row*NBIN + tid];
  __syncthreads();
  if (tid == 0) {
    int rem = k; int b = 0;
    for (int i = 255; i >= 0; i--) { if ((int)sb[i] >= rem) { b = i; break; } rem -= (int)sb[i]; }
    s_prefix = (unsigned)b; s_rem = rem;
  }
  __syncthreads();
  for (int L = 1; L < 4; L++) {
    sb[tid] = 0u; __syncthreads();
    const unsigned pref = s_prefix;
    const int shp = 32 - 8*L, shb = 24 - 8*L;
    for (int it = 0; it < NITER; it++) {
      float4 v = *(const float4*)(xr + it*TILE + tid*VEC);
      float e[4] = {v.x, v.y, v.z, v.w};
#pragma unroll
      for (int j = 0; j < 4; j++) {
        unsigned key = fkey(e[j]);
        if ((key >> shp) == pref) atomicAdd(&sb[(key >> shb) & 255], 1u);
      }
    }
    __syncthreads();
    if (tid == 0) {
      int rem = s_rem; int b = 0;
      for (int i = 255; i >= 0; i--) { if ((int)sb[i] >= rem) { b = i; break; } rem -= (int)sb[i]; }
      s_prefix = (s_prefix << 8) | (unsigned)b; s_rem = rem;
    }
    __syncthreads();
  }
  const unsigned key_k = s_prefix;

  // ---- weighted histogram over survivors: Z and top-p boundary ----
  swb[tid] = 0.f; __syncthreads();
  for (int it = 0; it < NITER; it++) {
    float4 v = *(const float4*)(xr + it*TILE + tid*VEC);
    float e[4] = {v.x, v.y, v.z, v.w};
#pragma unroll
    for (int j = 0; j < 4; j++) {
      unsigned key = fkey(e[j]);
      if (key >= key_k) atomicAdd(&swb[key >> 24], __expf(e[j] - m));
    }
  }
  __syncthreads();
  const float Z = block_sum(swb[tid], s_red);        // wmma-backed reduction
  const float target = top_ps[row]*Z;
  if (tid == 0) {
    float acc = 0.f; int b = -1;
    for (int i = 255; i >= 0; i--) { if (acc + swb[i] > target) { b = i; break; } acc += swb[i]; }
    s_hasP = (b >= 0);
    s_prefix = (b >= 0) ? (unsigned)b : 0u;
    s_trem = target - acc;
  }
  __syncthreads();
  unsigned key_p = 0u;
  if (s_hasP) {
    for (int L = 1; L < 4; L++) {
      swb[tid] = 0.f; __syncthreads();
      const unsigned pref = s_prefix;
      const int shp = 32 - 8*L, shb = 24 - 8*L;
      for (int it = 0; it < NITER; it++) {
        float4 v = *(const float4*)(xr + it*TILE + tid*VEC);
        float e[4] = {v.x, v.y, v.z, v.w};
#pragma unroll
        for (int j = 0; j < 4; j++) {
          unsigned key = fkey(e[j]);
          if (key >= key_k && (key >> shp) == pref)
            atomicAdd(&swb[(key >> shb) & 255], __expf(e[j] - m));
        }
      }
      __syncthreads();
      if (tid == 0) {
        float acc = 0.f, tr = s_trem; int b = 0;
        for (int i = 255; i >= 0; i--) { if (acc + swb[i] > tr) { b = i; break; } acc += swb[i]; }
        s_prefix = (s_prefix << 8) | (unsigned)b; s_trem = tr - acc;
      }
      __syncthreads();
    }
    key_p = s_prefix;   // first (largest) masked-by-top-p key; keep key > key_p
  }

  // top-a: p >= top_a/Z^2  <=>  x >= m + log(top_a) - log(Z)
  const float ta = top_as[row];
  const float xa = (ta > 0.f) ? (m + __logf(ta) - __logf(Z)) : -INFINITY;
  const unsigned key_m = fkey(m);
  const bool row0 = (row == 0);      // faithful mask[0,0]=False

  // Z' over final surviving set
  float accz = 0.f;
  for (int it = 0; it < NITER; it++) {
    float4 v = *(const float4*)(xr + it*TILE + tid*VEC);
    float e[4] = {v.x, v.y, v.z, v.w};
#pragma unroll
    for (int j = 0; j < 4; j++) {
      unsigned key = fkey(e[j]);
      bool keep = (key >= key_k) && (key > key_p) && (e[j] >= xa);
      if (row0 && key == key_m) keep = true;
      if (keep) accz += __expf(e[j] - m);
    }
  }
  const float Zp = block_sum(accz, s_red);           // wmma-backed reduction
  if (tid == 0) {
    float* r = ws_res + row*8;
    r[0] = m; r[1] = Zp;
    r[2] = __uint_as_float(key_k);
    r[3] = __uint_as_float(key_p);
    r[4] = xa;
  }
}

// ---------------- K3: elementwise finalize (in-place on d_out) --------------
__global__ __launch_bounds__(TPB) void k3_final(float* __restrict__ xio,
                                                const float* __restrict__ ws_res)
{
  const int row = blockIdx.y;
  const float* r = ws_res + row*8;
  const float m   = r[0];
  const float invZ = 1.0f / r[1];
  const unsigned key_k = __float_as_uint(r[2]);
  const unsigned key_p = __float_as_uint(r[3]);
  const float xa  = r[4];
  const unsigned key_m = fkey(m);
  const bool row0 = (row == 0);
  const size_t base = (size_t)row*VOCAB + (size_t)blockIdx.x*TILE + threadIdx.x*VEC;
  float4 v = *(float4*)(xio + base);
  float e[4] = {v.x, v.y, v.z, v.w};
#pragma unroll
  for (int j = 0; j < 4; j++) {
    unsigned key = fkey(e[j]);
    bool keep = (key >= key_k) && (key > key_p) && (e[j] >= xa);
    if (row0 && key == key_m) keep = true;
    e[j] = keep ? __expf(e[j] - m)*invZ : 0.0f;
  }
  v.x = e[0]; v.y = e[1]; v.z = e[2]; v.w = e[3];
  *(float4*)(xio + base) = v;
}

extern "C" void kernel_launch(void* const* d_in, const int* in_sizes, int n_in,
                              void* d_out, int out_size, void* d_ws, size_t ws_size,
                              hipStream_t stream)
{
  const float* logits    = (const float*)d_in[0];
  const int*   toks      = (const int*)  d_in[1];
  const float* presence  = (const float*)d_in[2];
  const float* frequency = (const float*)d_in[3];
  const float* temps     = (const float*)d_in[4];
  const float* top_ps    = (const float*)d_in[5];
  const int*   top_ks    = (const int*)  d_in[6];
  const float* top_as    = (const float*)d_in[7];
  float* out = (float*)d_out;

  float*    ws_max  = (float*)d_ws;                       // NSEQ
  unsigned* ws_hist = (unsigned*)(ws_max + NSEQ);         // NSEQ*256
  float*    ws_res  = (float*)(ws_hist + NSEQ*NBIN);      // NSEQ*8

  k1_penalize<<<NSEQ, TPB, 0, stream>>>(logits, toks, presence, frequency, temps,
                                        out, ws_max, ws_hist);
  k2_select<<<NSEQ, TPB, 0, stream>>>(out, ws_max, ws_hist, top_ks, top_ps, top_as,
                                      ws_res);
  k3_final<<<dim3(NITER, NSEQ), TPB, 0, stream>>>(out, ws_res);
}